// nchwBRA_76845554860108
// MI455X (gfx1250) — compile-verified
//
#include <hip/hip_runtime.h>
#include <math.h>

typedef float v2f __attribute__((ext_vector_type(2)));
typedef float v8f __attribute__((ext_vector_type(8)));

#define Bn 8
#define Cch 256
#define Hh 52
#define Ww 52
#define HP 56
#define WP 56
#define PIX (HP*WP)      // 3136 padded pixels
#define NPIX (Hh*Ww)     // 2704 real pixels
#define NREG 64
#define TK 4
#define SCL 0.0625f      // 256^-0.5

// workspace layout (float offsets)
#define SZ_PAD   (Bn*Cch*PIX)          // 6,422,528
#define OFF_QPAD 0
#define OFF_KPAD (OFF_QPAD + SZ_PAD)
#define SZ_R     (Bn*Cch*NREG)
#define OFF_QR   (OFF_KPAD + SZ_PAD)
#define OFF_KR   (OFF_QR + SZ_R)
#define SZ_AR    (Bn*NREG*NREG)
#define OFF_AR   (OFF_KR + SZ_R)
#define SZ_VALS  (Bn*NREG*TK*49)
#define OFF_VALS (OFF_AR + SZ_AR)
#define OFF_IDX  (OFF_VALS + SZ_VALS)  // ints live here

// ---------------------------------------------------------------------------
// Kernel 1: projection GEMM  out[o,p] = sum_c W[o,c] * X[c,p] + bias[o]
// f32 WMMA 16x16x4, one 16x16 tile per wave, K loop over 256 in steps of 4.
// Output scattered into zero-padded (b, o, 56x56) layout.
// ---------------------------------------------------------------------------
__global__ __launch_bounds__(256)
void proj_wmma(const float* __restrict__ X, const float* __restrict__ W,
               const float* __restrict__ bias, float* __restrict__ OUT)
{
    const int lane = threadIdx.x & 31;
    const int wave = threadIdx.x >> 5;
    const int tile = blockIdx.x * 8 + wave;          // 21632 tiles total
    const int b    = tile / (16 * 169);
    const int rem  = tile % (16 * 169);
    const int ot   = rem / 169;                      // output-row tile (16 rows)
    const int pt   = rem % 169;                      // pixel tile (16 cols)
    const int m    = lane & 15;
    const int koff = (lane >> 4) << 1;               // 0 or 2
    const int o0   = ot * 16, p0 = pt * 16;

    const float* Xb = X + (size_t)b * Cch * NPIX;

    v8f acc = {};
    for (int c0 = 0; c0 < Cch; c0 += 4) {
        v2f a, bf;
        const float* wp = W + (size_t)(o0 + m) * Cch + c0 + koff;
        a.x = wp[0];
        a.y = wp[1];
        const float* xp = Xb + (size_t)(c0 + koff) * NPIX + p0 + m;
        bf.x = xp[0];
        bf.y = xp[NPIX];
        __builtin_prefetch(xp + 4 * NPIX, 0, 1);     // global_prefetch_b8 next K-slice
        acc = __builtin_amdgcn_wmma_f32_16x16x4_f32(
                  false, a, false, bf, (short)0, acc, false, false);
    }

    const int p  = p0 + m;
    const int pp = (p / Ww) * WP + (p % Ww);         // padded pixel index
    const int ob = o0 + ((lane >> 4) << 3);
#pragma unroll
    for (int g = 0; g < 8; ++g) {
        const int o = ob + g;
        OUT[((size_t)b * Cch + o) * PIX + pp] = acc[g] + bias[o];
    }
}

// ---------------------------------------------------------------------------
// Kernel 2: ceil-mode 7x7 avg pool over padded buffer -> (b, c, 64)
// ---------------------------------------------------------------------------
__global__ __launch_bounds__(256)
void avgpool7(const float* __restrict__ P, float* __restrict__ R)
{
    const int i = blockIdx.x * blockDim.x + threadIdx.x;
    if (i >= Bn * Cch * NREG) return;
    const int r  = i & 63;
    const int bc = i >> 6;
    const int y = r >> 3, x = r & 7;
    const float* s = P + (size_t)bc * PIX + (y * 7) * WP + x * 7;
    float acc = 0.f;
#pragma unroll
    for (int ii = 0; ii < 7; ++ii)
#pragma unroll
        for (int jj = 0; jj < 7; ++jj)
            acc += s[ii * WP + jj];
    const float rc = (y == 7) ? 3.f : 7.f;
    const float cc = (x == 7) ? 3.f : 7.f;
    R[(size_t)bc * NREG + r] = acc / (rc * cc);
}

// ---------------------------------------------------------------------------
// Kernel 3: a_r[b,r1,r2] = sum_c q_r[b,c,r1]*k_r[b,c,r2]  (unscaled), WMMA f32
// 16 tiles per batch, 128 tiles = 16 blocks of 8 waves.
// ---------------------------------------------------------------------------
__global__ __launch_bounds__(256)
void region_affinity(const float* __restrict__ QR, const float* __restrict__ KR,
                     float* __restrict__ AR)
{
    const int lane = threadIdx.x & 31;
    const int wave = threadIdx.x >> 5;
    const int tile = blockIdx.x * 8 + wave;          // 128 tiles
    const int b  = tile >> 4;
    const int mt = (tile >> 2) & 3;
    const int nt = tile & 3;
    const int m    = lane & 15;
    const int koff = (lane >> 4) << 1;
    const float* Qb = QR + (size_t)b * Cch * NREG;
    const float* Kb = KR + (size_t)b * Cch * NREG;

    v8f acc = {};
    for (int c0 = 0; c0 < Cch; c0 += 4) {
        v2f a, bf;
        a.x  = Qb[(c0 + koff)     * NREG + mt * 16 + m];
        a.y  = Qb[(c0 + koff + 1) * NREG + mt * 16 + m];
        bf.x = Kb[(c0 + koff)     * NREG + nt * 16 + m];
        bf.y = Kb[(c0 + koff + 1) * NREG + nt * 16 + m];
        acc = __builtin_amdgcn_wmma_f32_16x16x4_f32(
                  false, a, false, bf, (short)0, acc, false, false);
    }
    const int n  = nt * 16 + m;
    const int mb = mt * 16 + ((lane >> 4) << 3);
#pragma unroll
    for (int g = 0; g < 8; ++g)
        AR[((size_t)b * NREG + mb + g) * NREG + n] = acc[g];
}

// ---------------------------------------------------------------------------
// Kernel 4: top-4 indices per a_r row (stable: ties -> lowest index first)
// ---------------------------------------------------------------------------
__global__ __launch_bounds__(256)
void topk4(const float* __restrict__ AR, int* __restrict__ IDX)
{
    const int i = blockIdx.x * blockDim.x + threadIdx.x;  // Bn*NREG = 512
    if (i >= Bn * NREG) return;
    const float* row = AR + (size_t)i * NREG;
    float v[NREG];
#pragma unroll
    for (int j = 0; j < NREG; ++j) v[j] = row[j];
    for (int s = 0; s < TK; ++s) {
        int best = 0; float bv = v[0];
        for (int j = 1; j < NREG; ++j)
            if (v[j] > bv) { bv = v[j]; best = j; }
        IDX[i * TK + s] = best;
        v[best] = -INFINITY;
    }
}

// ---------------------------------------------------------------------------
// Kernel 5: vals[b,r,j,t] = SCL * <q_seq[b,r,j,:], k_seq[b, idx[b,r,3], t, :]>
// j in [0,4), t in [0,49).  One block per (b,r).
// ---------------------------------------------------------------------------
__global__ __launch_bounds__(256)
void attn_vals(const float* __restrict__ QP, const float* __restrict__ KP,
               const int* __restrict__ IDX, float* __restrict__ VALS)
{
    const int br = blockIdx.x;                       // Bn*NREG
    const int b = br >> 6, r = br & 63;
    const int tid = threadIdx.x;
    if (tid >= TK * 49) return;
    const int j = tid / 49, t = tid % 49;
    const int rh = r >> 3, rw = r & 7;
    const int kreg = IDX[(br << 2) + 3];
    const int krh = kreg >> 3, krw = kreg & 7;
    const int pq = (rh * 7) * WP + rw * 7 + j;               // q in-region pos j (<7)
    const int pk = (krh * 7 + t / 7) * WP + krw * 7 + (t % 7);
    const float* qb = QP + (size_t)b * Cch * PIX;
    const float* kb = KP + (size_t)b * Cch * PIX;
    float acc = 0.f;
    for (int d = 0; d < Cch; ++d)
        acc += qb[(size_t)d * PIX + pq] * kb[(size_t)d * PIX + pk];
    VALS[(size_t)br * (TK * 49) + tid] = acc * SCL;
}

// ---------------------------------------------------------------------------
// Kernel 6: the 234MB output fill.  One block per (b,h1,w1); a_r column and
// vals row staged in LDS; non-temporal coalesced stores (output > L2).
// out[b,h1,w1,0,h2,w2] = SCL*a_r[b, regA(pb), regB(pa)]  (pa=r*49+q, pb=k*49+t
// re-split as 56x56 coords), overridden by vals when q<4 && k==idx[b,r,q].
// ---------------------------------------------------------------------------
__global__ __launch_bounds__(256)
void fill_out(const float* __restrict__ AR, const int* __restrict__ IDX,
              const float* __restrict__ VALS, float* __restrict__ OUT)
{
    __shared__ float sA[NREG];
    __shared__ float sV[49];
    __shared__ int   sK;
    const int blk = blockIdx.x;                      // Bn*Hh*Ww = 21632
    const int b  = blk / (Hh * Ww);
    const int hw = blk % (Hh * Ww);
    const int h1 = hw / Ww, w1 = hw % Ww;
    const int r = (h1 / 7) * 8 + (w1 / 7);
    const int q = (h1 % 7) * 7 + (w1 % 7);
    const int pa   = r * 49 + q;
    const int regB = ((pa / WP) / 7) * 8 + ((pa % WP) / 7);
    const int tid = threadIdx.x;

    if (tid < NREG)
        sA[tid] = AR[((size_t)b * NREG + tid) * NREG + regB] * SCL;
    if (tid == 0)
        sK = (q < TK) ? IDX[((b * NREG + r) << 2) + q] : -1;
    if (q < TK && tid < 49)
        sV[tid] = VALS[(((size_t)(b * NREG + r)) * TK + q) * 49 + tid];
    __syncthreads();

    float* outRow = OUT + (size_t)blk * (Hh * Ww);
    const int kidx = sK;
    for (int o = tid; o < Hh * Ww; o += 256) {
        const int h2 = o / Ww, w2 = o % Ww;
        const int k = (h2 / 7) * 8 + (w2 / 7);
        const int t = (h2 % 7) * 7 + (w2 % 7);
        const int pb   = k * 49 + t;
        const int regA = ((pb / WP) / 7) * 8 + ((pb % WP) / 7);
        float v = sA[regA];
        if (k == kidx) v = sV[t];
        __builtin_nontemporal_store(v, &outRow[o]);
    }
}

// ---------------------------------------------------------------------------
extern "C" void kernel_launch(void* const* d_in, const int* in_sizes, int n_in,
                              void* d_out, int out_size, void* d_ws, size_t ws_size,
                              hipStream_t stream)
{
    const float* x  = (const float*)d_in[0];
    const float* y  = (const float*)d_in[1];
    const float* qw = (const float*)d_in[2];
    const float* qb = (const float*)d_in[3];
    const float* kw = (const float*)d_in[4];
    const float* kb = (const float*)d_in[5];

    float* ws   = (float*)d_ws;
    float* qpad = ws + OFF_QPAD;
    float* kpad = ws + OFF_KPAD;
    float* qr   = ws + OFF_QR;
    float* kr   = ws + OFF_KR;
    float* ar   = ws + OFF_AR;
    float* vals = ws + OFF_VALS;
    int*   idx  = (int*)(ws + OFF_IDX);

    // zero padded projection buffers (pad region must be 0 for pool + seq)
    hipMemsetAsync(qpad, 0, (size_t)2 * SZ_PAD * sizeof(float), stream);

    proj_wmma<<<2704, 256, 0, stream>>>(x, qw, qb, qpad);
    proj_wmma<<<2704, 256, 0, stream>>>(y, kw, kb, kpad);

    const int npool = Bn * Cch * NREG;
    avgpool7<<<(npool + 255) / 256, 256, 0, stream>>>(qpad, qr);
    avgpool7<<<(npool + 255) / 256, 256, 0, stream>>>(kpad, kr);

    region_affinity<<<16, 256, 0, stream>>>(qr, kr, ar);
    topk4<<<2, 256, 0, stream>>>(ar, idx);
    attn_vals<<<Bn * NREG, 256, 0, stream>>>(qpad, kpad, idx, vals);
    fill_out<<<Bn * Hh * Ww, 256, 0, stream>>>(ar, idx, vals, (float*)d_out);
}